// MyModel_85864986182007
// MI455X (gfx1250) — compile-verified
//
#include <hip/hip_runtime.h>
#include <math.h>

// ---------------- constants ----------------
#define Bsz 16
#define Tsz 12
#define Nsz 1024
#define Esz 64
#define Gsz 64
#define NBANDS 16
#define NFREQ 5
#define TOPK 10
#define OUT_T 12

typedef float v2f __attribute__((ext_vector_type(2)));
typedef float v8f __attribute__((ext_vector_type(8)));

__device__ __forceinline__ v8f wmma_f32_16x16x4(v2f a, v2f b, v8f c) {
  return __builtin_amdgcn_wmma_f32_16x16x4_f32(false, a, false, b, (short)0, c,
                                               false, false);
}

__device__ __forceinline__ float sigmoidf_(float x) {
  return 1.0f / (1.0f + expf(-x));
}

// ---------------- 1. spectral features: fn [B*N, 16] ----------------
__global__ void spec_kernel(const float* __restrict__ flow,
                            const float* __restrict__ band_W,
                            float* __restrict__ fnbuf) {
  int tid = blockIdx.x * blockDim.x + threadIdx.x;  // B*N = 16384
  if (tid >= Bsz * Nsz) return;
  int b = tid >> 10, n = tid & 1023;
  float xv[12];
#pragma unroll
  for (int t = 0; t < Tsz; ++t) xv[t] = flow[((b * Tsz + t) << 10) + n];

  const float C8[8] = {1.f, 0.70710678f,  0.f, -0.70710678f,
                       -1.f, -0.70710678f, 0.f, 0.70710678f};
  const float S8[8] = {0.f, 0.70710678f,  1.f, 0.70710678f,
                       0.f, -0.70710678f, -1.f, -0.70710678f};
  float win[8];
#pragma unroll
  for (int t = 0; t < 8; ++t) win[t] = 0.5f * (1.f - C8[t]);

  float pw[NFREQ] = {0.f, 0.f, 0.f, 0.f, 0.f};
#pragma unroll
  for (int s = 0; s < 2; ++s) {  // Welch segments at t=0 and t=4
    int st = s * 4;
    float y[8];
#pragma unroll
    for (int t = 0; t < 8; ++t) y[t] = win[t] * xv[st + t];
#pragma unroll
    for (int k = 0; k < NFREQ; ++k) {
      float re = 0.f, im = 0.f;
#pragma unroll
      for (int t = 0; t < 8; ++t) {
        int m = (k * t) & 7;
        re += y[t] * C8[m];
        im -= y[t] * S8[m];
      }
      pw[k] += 0.5f * (re * re + im * im);  // mean over 2 segments
    }
  }
  float f5[NFREQ];
#pragma unroll
  for (int k = 0; k < NFREQ; ++k) f5[k] = logf(fmaxf(pw[k], 1e-8f));

  float f16a[NBANDS];
  float mu = 0.f;
#pragma unroll
  for (int j = 0; j < NBANDS; ++j) {
    float v = 0.f;
#pragma unroll
    for (int k = 0; k < NFREQ; ++k) v += f5[k] * band_W[j * NFREQ + k];
    f16a[j] = v;
    mu += v;
  }
  mu *= (1.f / NBANDS);
  float var = 0.f;
#pragma unroll
  for (int j = 0; j < NBANDS; ++j) {
    float d = f16a[j] - mu;
    var += d * d;
  }
  var *= (1.f / NBANDS);
  float rstd = 1.f / sqrtf(var + 1e-5f);
  float nrm = 0.f;
#pragma unroll
  for (int j = 0; j < NBANDS; ++j) {
    f16a[j] = (f16a[j] - mu) * rstd;
    nrm += f16a[j] * f16a[j];
  }
  float invn = 1.f / fmaxf(sqrtf(nrm), 1e-8f);
#pragma unroll
  for (int j = 0; j < NBANDS; ++j)
    fnbuf[(size_t)tid * NBANDS + j] = f16a[j] * invn;
}

// ---------------- 2. static adjacency + symmetrized static ----------------
__global__ void stat_kernel(const float* __restrict__ MS,
                            const float* __restrict__ noise,
                            float* __restrict__ out_static,
                            float* __restrict__ asym) {
  int tid = blockIdx.x * blockDim.x + threadIdx.x;  // N*N
  if (tid >= Nsz * Nsz) return;
  int i = tid >> 10, j = tid & 1023;
  float a = fminf(fmaxf(MS[tid] + 0.05f * noise[tid], 0.f), 1.f);
  int tji = (j << 10) + i;
  float at = fminf(fmaxf(MS[tji] + 0.05f * noise[tji], 0.f), 1.f);
  out_static[tid] = a;
  asym[tid] = 0.5f * (a + at);
}

// ---------------- 3. sim logits via f32 WMMA ----------------
// one wave -> 16x16 tile of fn @ fn^T / TEMP, written to A_dyn region
__global__ __launch_bounds__(128) void sim_kernel(const float* __restrict__ fn,
                                                  float* __restrict__ dyn) {
  unsigned t = threadIdx.x;
  unsigned lane = t & 31u, w = t >> 5;
  unsigned gw = blockIdx.x * 4u + w;  // 65536 waves
  unsigned b = gw >> 12;
  unsigned rem = gw & 4095u;
  unsigned i0 = (rem >> 6) * 16u, j0 = (rem & 63u) * 16u;
  unsigned half = lane >> 4, lr = lane & 15u;
  const float* fb = fn + (size_t)b * Nsz * NBANDS;

  v8f c;
#pragma unroll
  for (int r = 0; r < 8; ++r) c[r] = 0.f;

#pragma unroll
  for (int k0 = 0; k0 < 16; k0 += 4) {
    int kk = k0 + (int)half * 2;
    v2f a, bb;
    a.x = fb[(i0 + lr) * NBANDS + kk];
    a.y = fb[(i0 + lr) * NBANDS + kk + 1];
    bb.x = fb[(j0 + lr) * NBANDS + kk];
    bb.y = fb[(j0 + lr) * NBANDS + kk + 1];
    c = wmma_f32_16x16x4(a, bb, c);
  }
  float* d = dyn + (size_t)b * Nsz * Nsz;
#pragma unroll
  for (int r = 0; r < 8; ++r) {
    unsigned row = i0 + r + half * 8;
    d[(size_t)row * Nsz + j0 + lr] = c[r] * (1.0f / 0.07f);
  }
}

// ---------------- 4. row softmax + zero-diag + top-10 mask (in place) -----
// 256 threads, each owns one float4 of the 1024-wide row
__global__ __launch_bounds__(256) void soft_kernel(float* __restrict__ dyn) {
  __shared__ float rowv[1024];
  __shared__ float red[256];
  __shared__ int ridx[256];
  __shared__ unsigned char msk[1024];
  unsigned t = threadIdx.x;
  unsigned bn = blockIdx.x;  // b*1024 + i
  unsigned i = bn & 1023u;
  float* row = dyn + (size_t)bn * Nsz;

  float4 av = ((const float4*)row)[t];  // j = 4t..4t+3, single b128 load
  ((unsigned int*)msk)[t] = 0u;

  // row max from registers
  float m = fmaxf(fmaxf(av.x, av.y), fmaxf(av.z, av.w));
  red[t] = m;
  __syncthreads();
  for (int o = 128; o > 0; o >>= 1) {
    if (t < o) red[t] = fmaxf(red[t], red[t + o]);
    __syncthreads();
  }
  m = red[0];
  __syncthreads();
  // exp-sum from registers
  float e0 = expf(av.x - m), e1 = expf(av.y - m);
  float e2 = expf(av.z - m), e3 = expf(av.w - m);
  red[t] = (e0 + e1) + (e2 + e3);
  __syncthreads();
  for (int o = 128; o > 0; o >>= 1) {
    if (t < o) red[t] += red[t + o];
    __syncthreads();
  }
  float inv = 1.f / red[0];
  __syncthreads();
  // normalized values, zero the diagonal AFTER softmax
  unsigned jb = t * 4u;
  rowv[jb + 0] = (jb + 0 == i) ? 0.f : e0 * inv;
  rowv[jb + 1] = (jb + 1 == i) ? 0.f : e1 * inv;
  rowv[jb + 2] = (jb + 2 == i) ? 0.f : e2 * inv;
  rowv[jb + 3] = (jb + 3 == i) ? 0.f : e3 * inv;
  __syncthreads();
  // iterative top-10, lowest-index tie-break (matches lax.top_k)
  for (int it = 0; it < TOPK; ++it) {
    float bv = -1.f;
    int bj = 0;
#pragma unroll
    for (int u = 0; u < 4; ++u) {  // ascending j -> earliest kept on ties
      int j = jb + u;
      float v = msk[j] ? -1.f : rowv[j];
      if (v > bv) { bv = v; bj = j; }
    }
    red[t] = bv;
    ridx[t] = bj;
    __syncthreads();
    for (int o = 128; o > 0; o >>= 1) {
      if (t < o) {
        float ov = red[t + o];
        int oi = ridx[t + o];
        if (ov > red[t] || (ov == red[t] && oi < ridx[t])) {
          red[t] = ov;
          ridx[t] = oi;
        }
      }
      __syncthreads();
    }
    if (t == 0) msk[ridx[0]] = 1;
    __syncthreads();
  }
  float4 ov;
  ov.x = msk[jb + 0] ? rowv[jb + 0] : 0.f;
  ov.y = msk[jb + 1] ? rowv[jb + 1] : 0.f;
  ov.z = msk[jb + 2] ? rowv[jb + 2] : 0.f;
  ov.w = msk[jb + 3] ? rowv[jb + 3] : 0.f;
  ((float4*)row)[t] = ov;  // single b128 store
}

// ---------------- 5. symmetrize + fuse (tile-pair, in place) -------------
__global__ __launch_bounds__(256) void sym_kernel(float* __restrict__ dyn,
                                                  float* __restrict__ fused,
                                                  const float* __restrict__ asym,
                                                  const float* __restrict__ fa) {
  __shared__ float T1[32][33], T2[32][33], S1[32][33];
  int b = blockIdx.x, ti = blockIdx.y, tj = blockIdx.z;
  if (ti > tj) return;
  float alpha = sigmoidf_(fa[0]);
  int t = threadIdx.x, tx = t & 31, ty = t >> 5;
  float* P = dyn + (size_t)b * Nsz * Nsz;
  float* F = fused + (size_t)b * Nsz * Nsz;
  int r0 = ti * 32, c0 = tj * 32;
#pragma unroll
  for (int s = 0; s < 4; ++s) {
    int r = ty + s * 8;
    T1[r][tx] = P[(size_t)(r0 + r) * Nsz + c0 + tx];
    T2[r][tx] = P[(size_t)(c0 + r) * Nsz + r0 + tx];
    S1[r][tx] = asym[(size_t)(r0 + r) * Nsz + c0 + tx];
  }
  __syncthreads();
#pragma unroll
  for (int s = 0; s < 4; ++s) {
    int r = ty + s * 8;
    float d1 = 0.5f * (T1[r][tx] + T2[tx][r]);
    float f1 = alpha * d1 + (1.f - alpha) * S1[r][tx];
    P[(size_t)(r0 + r) * Nsz + c0 + tx] = d1;
    F[(size_t)(r0 + r) * Nsz + c0 + tx] = f1;
    if (ti != tj) {
      float d2 = 0.5f * (T2[r][tx] + T1[tx][r]);
      float f2 = alpha * d2 + (1.f - alpha) * S1[tx][r];  // asym symmetric
      P[(size_t)(c0 + r) * Nsz + r0 + tx] = d2;
      F[(size_t)(c0 + r) * Nsz + r0 + tx] = f2;
    }
  }
}

// ---------------- 6. degree -> dinv (deterministic, no atomics) ----------
__global__ __launch_bounds__(256) void deg_kernel(const float* __restrict__ fused,
                                                  float* __restrict__ dinv) {
  __shared__ float red[256];
  int bn = blockIdx.x, t = threadIdx.x;
  const float* row = fused + (size_t)bn * Nsz;
  float4 av = ((const float4*)row)[t];  // one b128 per thread covers the row
  red[t] = (fmaxf(av.x, 0.f) + fmaxf(av.y, 0.f)) +
           (fmaxf(av.z, 0.f) + fmaxf(av.w, 0.f));
  __syncthreads();
  for (int o = 128; o > 0; o >>= 1) {
    if (t < o) red[t] += red[t + o];
    __syncthreads();
  }
  if (t == 0) {
    float d = fmaxf(1.f + red[0], 1e-6f);  // +1 for identity
    dinv[bn] = 1.f / sqrtf(d);
  }
}

// ---------------- 7. x extraction + xd = dinv*x --------------------------
__global__ void xprep_kernel(const float* __restrict__ flow,
                             const float* __restrict__ dinv,
                             float* __restrict__ x, float* __restrict__ xd) {
  int tid = blockIdx.x * blockDim.x + threadIdx.x;
  if (tid >= Bsz * Nsz) return;
  int b = tid >> 10, n = tid & 1023;
  float di = dinv[tid];
#pragma unroll
  for (int t = 0; t < Tsz; ++t) {
    float v = flow[((b * Tsz + t) << 10) + n];
    x[(size_t)tid * Tsz + t] = v;
    xd[(size_t)tid * Tsz + t] = di * v;
  }
}

// ---------------- 8. propagation: y = A_hat @ v --------------------------
// y[i,f] = dinv[i]*(sum_j relu(Af[i,j])*vd[j,f]) + dinv[i]^2*raw[i,f]
__global__ __launch_bounds__(256) void prop_kernel(
    const float* __restrict__ fused, const float* __restrict__ dinv,
    const float* __restrict__ vd, const float* __restrict__ raw,
    float* __restrict__ y, float* __restrict__ yd) {
  __shared__ float vds[Nsz * Tsz];  // 48 KB (well within 320KB WGP LDS)
  __shared__ float red[256];
  __shared__ float res[Tsz];
  int t = threadIdx.x;
  int bn = blockIdx.x, b = bn >> 10;
  const float* row = fused + (size_t)bn * Nsz;
  __builtin_prefetch(row + 4 * t, 0, 3);  // global_prefetch_b8: start row fetch
  // stage this batch's vd (dinv-scaled features) into LDS, b128 loads
  const float4* vb4 = (const float4*)(vd + (size_t)b * Nsz * Tsz);
  float4* vds4 = (float4*)vds;
  for (int idx = t; idx < (Nsz * Tsz) / 4; idx += 256) vds4[idx] = vb4[idx];
  __syncthreads();

  float4 av = ((const float4*)row)[t];  // adjacency row chunk j=4t..4t+3
  float acc[Tsz];
#pragma unroll
  for (int f = 0; f < Tsz; ++f) acc[f] = 0.f;
  {
    float aa[4] = {fmaxf(av.x, 0.f), fmaxf(av.y, 0.f), fmaxf(av.z, 0.f),
                   fmaxf(av.w, 0.f)};
#pragma unroll
    for (int u = 0; u < 4; ++u) {
      int j = t * 4 + u;
#pragma unroll
      for (int f = 0; f < Tsz; ++f) acc[f] += aa[u] * vds[j * Tsz + f];
    }
  }
  for (int f = 0; f < Tsz; ++f) {
    red[t] = acc[f];
    __syncthreads();
    for (int o = 128; o > 0; o >>= 1) {
      if (t < o) red[t] += red[t + o];
      __syncthreads();
    }
    if (t == 0) res[f] = red[0];
    __syncthreads();
  }
  if (t < Tsz) {
    float di = dinv[bn];
    float v = di * res[t] + di * di * raw[(size_t)bn * Tsz + t];
    y[(size_t)bn * Tsz + t] = v;
    yd[(size_t)bn * Tsz + t] = di * v;
  }
}

// ---------------- 9. site encoder ----------------------------------------
__global__ void site_kernel(const float* __restrict__ site_emb,
                            const float* __restrict__ embed_W,
                            const float* __restrict__ embed_b,
                            float* __restrict__ site) {
  int tid = blockIdx.x * blockDim.x + threadIdx.x;  // N*E
  if (tid >= Nsz * Esz) return;
  int n = tid >> 6, e = tid & 63;
  float s = embed_b[e];
#pragma unroll
  for (int k = 0; k < 16; ++k) s += site_emb[n * 16 + k] * embed_W[k * Esz + e];
  site[tid] = s;
}

// ---------------- 10. station embedding + layernorm ----------------------
__global__ __launch_bounds__(64) void emb_kernel(
    const float* __restrict__ x, const float* __restrict__ site,
    const float* __restrict__ flow_W, const float* __restrict__ flow_b,
    const float* __restrict__ ln_w, const float* __restrict__ ln_b,
    float* __restrict__ emb) {
  __shared__ float red[64];
  int t = threadIdx.x;
  int bn = blockIdx.x, n = bn & 1023;
  float ff = flow_b[t];
#pragma unroll
  for (int tt = 0; tt < Tsz; ++tt)
    ff += x[(size_t)bn * Tsz + tt] * flow_W[tt * Esz + t];
  float sf = site[(size_t)n * Esz + t];
  float p = sigmoidf_(sf);
  float q = tanhf(ff);
  float v = p * q + (1.f - p) * sf;
  red[t] = v;
  __syncthreads();
  for (int o = 32; o > 0; o >>= 1) {
    if (t < o) red[t] += red[t + o];
    __syncthreads();
  }
  float mu = red[0] * (1.f / 64.f);
  __syncthreads();
  red[t] = v * v;
  __syncthreads();
  for (int o = 32; o > 0; o >>= 1) {
    if (t < o) red[t] += red[t + o];
    __syncthreads();
  }
  float var = red[0] * (1.f / 64.f) - mu * mu;
  float ev = (v - mu) / sqrtf(var + 1e-5f) * ln_w[t] + ln_b[t];
  emb[(size_t)bn * Esz + t] = ev;
}

// ---------------- 11. per-node GEMM [16384,2304]x[2304,64] via WMMA ------
// R[m, e*36+kf] = emb[m,e] * xg[m,kf] generated on the fly from LDS strips.
__global__ __launch_bounds__(128) void node_kernel(
    const float* __restrict__ emb, const float* __restrict__ xx,
    const float* __restrict__ y1, const float* __restrict__ y2,
    const float* __restrict__ Wp, const float* __restrict__ bias_pool,
    float* __restrict__ g) {
  __shared__ float emb_s[16 * 64];
  __shared__ float xg_s[16 * 36];
  int t = threadIdx.x;
  int node0 = blockIdx.x * 16;
  for (int idx = t; idx < 16 * 64; idx += 128)
    emb_s[idx] = emb[(size_t)node0 * 64 + idx];
  for (int idx = t; idx < 16 * 36; idx += 128) {
    int m = idx / 36, kf = idx - m * 36;
    int k = kf / 12, f = kf - k * 12;
    size_t p = (size_t)(node0 + m) * Tsz + f;
    float v = (k == 0) ? xx[p] : (k == 1 ? y1[p] : (2.f * y2[p] - xx[p]));
    xg_s[idx] = v;
  }
  __syncthreads();
  int lane = t & 31, wv = t >> 5;
  int half = lane >> 4, lr = lane & 15;
  int col = wv * 16 + lr;  // 4 waves cover the 64 output channels

  v8f c;
#pragma unroll
  for (int r = 0; r < 8; ++r) c[r] = 0.f;

  for (int kc = 0; kc < 576; ++kc) {  // K = 2304 = 576 * 4
    int idx0 = kc * 4 + half * 2;
    int idx1 = idx0 + 1;
    int e0 = idx0 / 36, kf0 = idx0 - e0 * 36;
    int e1 = idx1 / 36, kf1 = idx1 - e1 * 36;
    v2f a, bb;
    a.x = emb_s[lr * 64 + e0] * xg_s[lr * 36 + kf0];
    a.y = emb_s[lr * 64 + e1] * xg_s[lr * 36 + kf1];
    bb.x = Wp[(size_t)idx0 * 64 + col];
    bb.y = Wp[(size_t)idx1 * 64 + col];
    c = wmma_f32_16x16x4(a, bb, c);
  }
#pragma unroll
  for (int r = 0; r < 8; ++r) {
    int m = r + half * 8;
    float bia = 0.f;
    for (int e = 0; e < 64; ++e)
      bia += emb_s[m * 64 + e] * bias_pool[e * 64 + col];
    float gv = c[r] + bia;
    gv = 0.5f * gv * (1.f + erff(gv * 0.70710678118f));  // exact gelu
    g[(size_t)(node0 + m) * 64 + col] = gv;
  }
}

// ---------------- 12. gated decoder + final projection -------------------
__global__ __launch_bounds__(64) void dec_kernel(
    const float* __restrict__ g, const float* __restrict__ gate_W,
    const float* __restrict__ gate_b, const float* __restrict__ up_W,
    const float* __restrict__ up_b, const float* __restrict__ fc_W,
    const float* __restrict__ fc_b, float* __restrict__ out) {
  __shared__ float gs[64];
  __shared__ float hs[64];
  int t = threadIdx.x;
  int bn = blockIdx.x, b = bn >> 10, n = bn & 1023;
  gs[t] = g[(size_t)bn * 64 + t];
  __syncthreads();
  float s1 = gate_b[t], s2 = up_b[t];
  for (int i = 0; i < 64; ++i) {
    float gv = gs[i];
    s1 += gv * gate_W[i * 64 + t];
    s2 += gv * up_W[i * 64 + t];
  }
  float p = sigmoidf_(s1);
  float q = tanhf(s2);
  hs[t] = p * q + (1.f - p) * gs[t];
  __syncthreads();
  if (t < OUT_T) {
    float v = fc_b[t];
    for (int i = 0; i < 64; ++i) v += hs[i] * fc_W[i * OUT_T + t];
    out[((b * OUT_T + t) << 10) + n] = v;  // [B, OT, N, 1]
  }
}

// ---------------- host launch ----------------
extern "C" void kernel_launch(void* const* d_in, const int* in_sizes, int n_in,
                              void* d_out, int out_size, void* d_ws,
                              size_t ws_size, hipStream_t stream) {
  (void)in_sizes; (void)n_in; (void)out_size; (void)ws_size;
  const float* flow        = (const float*)d_in[0];
  const float* noise       = (const float*)d_in[1];
  const float* MS          = (const float*)d_in[2];
  // d_in[3] = feature_logits (F=1 -> softmax == 1.0, unused)
  const float* band_W      = (const float*)d_in[4];
  const float* site_emb    = (const float*)d_in[5];
  const float* flow_W      = (const float*)d_in[6];
  const float* flow_b      = (const float*)d_in[7];
  const float* embed_W     = (const float*)d_in[8];
  const float* embed_b     = (const float*)d_in[9];
  const float* ln_w        = (const float*)d_in[10];
  const float* ln_b        = (const float*)d_in[11];
  const float* weights_pool= (const float*)d_in[12];
  const float* bias_pool   = (const float*)d_in[13];
  const float* gate_W      = (const float*)d_in[14];
  const float* gate_b      = (const float*)d_in[15];
  const float* up_W        = (const float*)d_in[16];
  const float* up_b        = (const float*)d_in[17];
  const float* fc_W        = (const float*)d_in[18];
  const float* fc_b        = (const float*)d_in[19];
  const float* fusion_alpha= (const float*)d_in[20];

  float* out = (float*)d_out;
  float* out_main   = out;                                   // 196608
  float* out_static = out + 196608;                          // 1048576
  float* out_dyn    = out + 196608 + 1048576;                // 16777216
  float* out_fused  = out + 196608 + 1048576 + 16777216;     // 16777216

  // workspace layout (floats), ~18 MB total
  float* ws    = (float*)d_ws;
  float* fnbuf = ws;                         // 262144
  float* asym  = fnbuf + 262144;             // 1048576
  float* dinv  = asym + 1048576;             // 16384
  float* xbuf  = dinv + 16384;               // 196608
  float* xd    = xbuf + 196608;              // 196608
  float* y1    = xd + 196608;                // 196608
  float* y1d   = y1 + 196608;                // 196608
  float* y2    = y1d + 196608;               // 196608
  float* y2d   = y2 + 196608;                // 196608
  float* site  = y2d + 196608;               // 65536
  float* embb  = site + 65536;               // 1048576
  float* gbuf  = embb + 1048576;             // 1048576

  // 1. spectral node features
  spec_kernel<<<64, 256, 0, stream>>>(flow, band_W, fnbuf);
  // 2. static adjacency (+ symmetrized copy)
  stat_kernel<<<4096, 256, 0, stream>>>(MS, noise, out_static, asym);
  // 3. sim logits (WMMA f32 16x16x4), into A_dyn region
  sim_kernel<<<16384, 128, 0, stream>>>(fnbuf, out_dyn);
  // 4. softmax + zero diag + top-10 (in place)
  soft_kernel<<<16384, 256, 0, stream>>>(out_dyn);
  // 5. symmetrize + fuse
  {
    dim3 grid(Bsz, 32, 32);
    sym_kernel<<<grid, 256, 0, stream>>>(out_dyn, out_fused, asym, fusion_alpha);
  }
  // 6. degrees
  deg_kernel<<<16384, 256, 0, stream>>>(out_fused, dinv);
  // 7. x and xd
  xprep_kernel<<<64, 256, 0, stream>>>(flow, dinv, xbuf, xd);
  // 8. Chebyshev propagation: y1 = A_hat@x ; y2 = A_hat@y1
  prop_kernel<<<16384, 256, 0, stream>>>(out_fused, dinv, xd, xbuf, y1, y1d);
  prop_kernel<<<16384, 256, 0, stream>>>(out_fused, dinv, y1d, y1, y2, y2d);
  // 9-10. station encoder
  site_kernel<<<256, 256, 0, stream>>>(site_emb, embed_W, embed_b, site);
  emb_kernel<<<16384, 64, 0, stream>>>(xbuf, site, flow_W, flow_b, ln_w, ln_b,
                                       embb);
  // 11. per-node bilinear GEMM (WMMA) + bias + gelu
  node_kernel<<<1024, 128, 0, stream>>>(embb, xbuf, y1, y2, weights_pool,
                                        bias_pool, gbuf);
  // 12. gated decoder + output projection
  dec_kernel<<<16384, 64, 0, stream>>>(gbuf, gate_W, gate_b, up_W, up_b, fc_W,
                                       fc_b, out_main);
}